// PQMF_84018150245100
// MI455X (gfx1250) — compile-verified
//
#include <hip/hip_runtime.h>

// ---------------------------------------------------------------------------
// Fused PQMF analysis+synthesis for MI455X (gfx1250).
//   f32 WMMA 16x16x4 compute + Tensor Data Mover (TDM) global->LDS staging.
//
//   x   : (16, 524288) f32        qmf : (4, 65) f32        y : (16, 524288)
//
//   sub[b,k,s] = sum_t x_pad[b, 4s+t-32] * qmf[k,t]          (stride-4 conv)
//   y[b,n]     = sum_{k,t: (n+t-32)%4==0, 0<=s<131072}
//                  qmf[k,t] * sub[b,k,(n+t-32)/4]            (dilated conv)
//
// GEMM mapping: M = 16 batches, N = 16 sample positions, K = taps.
// Stage 1: A[b,u]=x[b,4s0-32+u], B_k[u,j]=qmf[k,u-4j], K=128 (32 wmma).
// Stage 2: A[b,(c,k)]=sub[b,k,nb/4+c-8], B[(c,k),j]=qmf[k,4c-j], K=80 (20 wmma).
//
// x tile staged by TDM with pad_enable: 4 pad dwords every 128 dwords ->
// row pitch 396 dwords; 396 % 64 == 12 makes batch-striped LDS reads
// bank-conflict-free. Pad offset is piecewise constant per task:
//   xc = 64h + 4c + off2 crosses a 128-dword boundary only at c==16, h odd,
// so the pad folds into the base pointer + one constant on the second half.
// ---------------------------------------------------------------------------

typedef float v2f __attribute__((ext_vector_type(2)));
typedef float v8f __attribute__((ext_vector_type(8)));
typedef unsigned int u32x4 __attribute__((ext_vector_type(4)));
typedef int i32x4 __attribute__((ext_vector_type(4)));
typedef int i32x8 __attribute__((ext_vector_type(8)));

#define LEN     524288
#define SLEN    (LEN / 4)       // 131072 subband samples
#define NBATCH  16
#define TILE_N  256             // output samples per workgroup
#define XCOLS   (TILE_N + 128)  // 384 x columns cached: [n0-64, n0+320)
#define XPITCH  396             // 384 + 3 TDM pads of 4 dwords
#define SUBCAP  82              // si in [0,80) used; padded

#if defined(__has_builtin)
#if __has_builtin(__builtin_amdgcn_tensor_load_to_lds) && \
    __has_builtin(__builtin_amdgcn_s_wait_tensorcnt)
#define USE_TDM 1
#endif
#endif
#ifndef USE_TDM
#define USE_TDM 0
#endif

// padded LDS column: insert 4 dwords after every 128 stored dwords (TDM pad)
__device__ __forceinline__ int xpad(int c) { return c + ((c >> 7) << 2); }

__global__ __launch_bounds__(256)
void pqmf_fused_kernel(const float* __restrict__ x,
                       const float* __restrict__ qmf,
                       float* __restrict__ y)
{
    __shared__ __align__(16) float x_s[NBATCH][XPITCH];      // x tile (TDM dest)
    __shared__ __align__(16) float sub_s[NBATCH][SUBCAP][4]; // sub tile, k innermost
    __shared__ __align__(16) float qmfKT[4][68];             // qmf[k][t]; 65..67 = 0
    __shared__ __align__(16) float qmfTK[66][4];             // qmf[t][k]; row 65 = 0

    const int  tid  = threadIdx.x;
    const int  lane = tid & 31;          // wave32
    const int  wave = tid >> 5;          // 8 waves
    const int  j    = lane & 15;
    const int  grp  = lane >> 4;         // 0: K=0,1   1: K=2,3
    const int  off2 = grp * 2;
    const long n0   = (long)blockIdx.x * TILE_N;

    // ---- qmf into LDS in both layouts, zero-padded ----
    for (int i = tid; i < 4 * 68; i += 256) {
        int k = i / 68, t = i % 68;
        qmfKT[k][t] = (t < 65) ? qmf[k * 65 + t] : 0.0f;
    }
    for (int i = tid; i < 66 * 4; i += 256) {
        int t = i >> 2, k = i & 3;
        qmfTK[t][k] = (t < 65) ? qmf[k * 65 + t] : 0.0f;
    }

    // ---- x tile [n0-64, n0+320) x 16 batches -> LDS ----
    const bool interior = (n0 >= 64) && (n0 + (XCOLS - 64) <= (long)LEN);
#if USE_TDM
    if (interior) {
        if (wave == 0) {
            // ---- Tensor DMA descriptor (D#), 2D tile: 384 cols x 16 rows ----
            const unsigned long long gaddr =
                (unsigned long long)(const void*)x + (unsigned long long)(n0 - 64) * 4ull;
            const unsigned int lds_addr = (unsigned int)(unsigned long long)(&x_s[0][0]);

            u32x4 g0;
            g0[0] = 1u;                                  // count=1, user mode, no gather
            g0[1] = lds_addr;                            // [63:32]  lds_addr
            g0[2] = (unsigned int)(gaddr & 0xFFFFFFFFull);           // global_addr lo
            g0[3] = (unsigned int)((gaddr >> 32) & 0x1FFFFFFull)     // global_addr[56:32]
                    | (2u << 30);                        // type = 2 ("image")

            i32x8 g1;
            g1[0] = (2 << 16)                            // data_size = 4 bytes
                  | (1 << 20)                            // pad_enable
                  | (6 << 22)                            // pad_interval: 128 dwords
                  | (3 << 25);                           // pad_amount: 4 dwords
            g1[1] = (int)((0x40000000u & 0xFFFFu) << 16);        // tensor_dim0 lo16 (huge)
            g1[2] = (int)(((0x40000000u >> 16) & 0xFFFFu)        // tensor_dim0 hi16
                  | (16u << 16));                        // tensor_dim1 = 16 rows (lo16)
            g1[3] = (int)(0u | ((unsigned)XCOLS << 16)); // tensor_dim1 hi=0, tile_dim0=384
            g1[4] = 16;                                  // tile_dim1 = 16, tile_dim2 = 0
            g1[5] = (int)LEN;                            // tensor_dim0_stride = 524288 (lo32)
            g1[6] = 0;                                   // stride hi, tensor_dim1_stride lo
            g1[7] = 0;

            i32x4 gz4 = {0, 0, 0, 0};                    // 2D: groups 2/3 unused
            i32x8 gz8 = {0, 0, 0, 0, 0, 0, 0, 0};

            __builtin_amdgcn_tensor_load_to_lds(g0, g1, gz4, gz4, gz8, 0);
            __builtin_amdgcn_s_wait_tensorcnt(0);

            // prefetch next tile (global_prefetch_b8), purely speculative
            long gn = n0 + TILE_N + lane * 64;
            if (gn < LEN) __builtin_prefetch(x + gn, 0, 1);
        }
    } else
#endif
    {
        // boundary tiles (and TDM-less builds): clamped scalar staging
        for (int rb = 0; rb < 2; ++rb) {
            const int b = wave * 2 + rb;
            const float* xrow = x + (long)b * LEN;
            for (int c = lane; c < XCOLS; c += 32) {
                long g = n0 - 64 + c;
                x_s[b][xpad(c)] =
                    ((unsigned long)g < (unsigned long)LEN) ? xrow[g] : 0.0f;
            }
        }
    }
    __syncthreads();

    // ================= Stage 1: analysis (stride-4 conv) ====================
    // tasks: (k in [0,4)) x (h in [0,5)) ; each yields sub[., k, 16 s-values]
    // k = task & 3 is wave-invariant (stride 8), so B fragments hoist per wave.
    const long s_start = n0 / 4 - 8;
    for (int task = wave; task < 20; task += 8) {
        const int k = task & 3;
        const int h = task >> 2;
        // pad folded into base: xpad(64h + q) = 64h + 4*(h>>1) + q  for q < split
        const float* abase = &x_s[j][64 * h + 4 * (h >> 1) + off2];
        const int    extra = (h & 1) * 4;   // second-half pad step (c >= 16)
        v8f acc = {};
        #pragma unroll
        for (int c = 0; c < 16; ++c) {
            v2f a = *(const v2f*)(abase + 4 * c);
            const int t0 = 4 * (c - j) + off2;
            v2f bf = *(const v2f*)&qmfKT[k][((unsigned)t0 <= 64u) ? t0 : 66];
            acc = __builtin_amdgcn_wmma_f32_16x16x4_f32(
                false, a, false, bf, (short)0, acc, false, false);
        }
        #pragma unroll
        for (int c = 16; c < 32; ++c) {
            v2f a = *(const v2f*)(abase + 4 * c + extra);
            const int t0 = 4 * (c - j) + off2;
            v2f bf = *(const v2f*)&qmfKT[k][((unsigned)t0 <= 64u) ? t0 : 66];
            acc = __builtin_amdgcn_wmma_f32_16x16x4_f32(
                false, a, false, bf, (short)0, acc, false, false);
        }
        // D layout: vgpr r -> (M = r + 8*grp, N = j); zero out-of-range s
        const int  si   = 16 * h + j;
        const long sg   = s_start + si;
        const bool sval = ((unsigned long)sg < (unsigned long)SLEN);
        #pragma unroll
        for (int r = 0; r < 8; ++r) {
            sub_s[r + grp * 8][si][k] = sval ? acc[r] : 0.0f;
        }
    }
    __syncthreads();

    // ================= Stage 2: synthesis (dilated conv) ====================
    // output block nb = n0 + 16g ; K = (c in [0,20)) x (k in [0,4))
    for (int g = wave; g < 16; g += 8) {
        const float* abase = &sub_s[j][4 * g][off2];
        v8f acc = {};
        #pragma unroll
        for (int c = 0; c < 20; ++c) {
            v2f a = *(const v2f*)(abase + 4 * c);
            const int t   = 4 * c - j;
            const int row = ((unsigned)t <= 64u) ? t : 65;
            v2f bf = *(const v2f*)&qmfTK[row][off2];
            acc = __builtin_amdgcn_wmma_f32_16x16x4_f32(
                false, a, false, bf, (short)0, acc, false, false);
        }
        const long nb = n0 + 16 * g;
        #pragma unroll
        for (int r = 0; r < 8; ++r) {
            const int b = r + grp * 8;
            y[(long)b * LEN + nb + j] = acc[r];
        }
    }
}

extern "C" void kernel_launch(void* const* d_in, const int* in_sizes, int n_in,
                              void* d_out, int out_size, void* d_ws, size_t ws_size,
                              hipStream_t stream) {
    const float* x   = (const float*)d_in[0];   // (16,1,524288) f32
    const float* qmf = (const float*)d_in[1];   // (4,65) f32
    float* y = (float*)d_out;                   // (16,1,524288) f32
    (void)in_sizes; (void)n_in; (void)out_size; (void)d_ws; (void)ws_size;

    dim3 grid(LEN / TILE_N);                    // 2048 workgroups
    dim3 block(256);                            // 8 wave32 waves
    pqmf_fused_kernel<<<grid, block, 0, stream>>>(x, qmf, y);
}